// cpu_iter_cal_grad_47090021433585
// MI455X (gfx1250) — compile-verified
//
#include <hip/hip_runtime.h>

#define NL   64
#define NOBS 400
#define NC   3000
#define DAMP 1.0f

#define TWO_PI_F 6.2831853071795864769f
#define LN01_F  -2.3025850929940456840f   // ln(0.1)

typedef float v2f __attribute__((ext_vector_type(2)));
typedef float v8f __attribute__((ext_vector_type(8)));

// ---------------------------------------------------------------------------
// helpers
// ---------------------------------------------------------------------------
__device__ __forceinline__ float fast_rcp(float x) {
  return __builtin_amdgcn_rcpf(x);
}
__device__ __forceinline__ float fast_sqrt(float x) {
  return __builtin_amdgcn_sqrtf(x);
}

// Order-preserving float<->uint encoding so integer atomicMax/Min give
// deterministic float extrema.
__device__ __forceinline__ unsigned enc_f(float f) {
  unsigned u = __float_as_uint(f);
  return (u & 0x80000000u) ? ~u : (u | 0x80000000u);
}
__device__ __forceinline__ float dec_f(unsigned u) {
  unsigned v = (u & 0x80000000u) ? (u & 0x7FFFFFFFu) : ~u;
  return __uint_as_float(v);
}

// ---------------------------------------------------------------------------
// Rayleigh dispersion determinant (Dunkin recursion), one (c, omega) pair.
// Layer-constant arrays are omega-independent (precomputed): d, 1/a, 1/b,
// 2*b^2, rho, 1/rho, 1/rho^2.  Index NL-1 = halfspace.
// ---------------------------------------------------------------------------
__device__ __forceinline__ float dltar4_dev(
    float cvel, float omega,
    const float* __restrict__ sD,    const float* __restrict__ sAinv,
    const float* __restrict__ sBinv, const float* __restrict__ sB2,
    const float* __restrict__ sRho,  const float* __restrict__ sRinv,
    const float* __restrict__ sR2inv)
{
  const float wvno  = omega / cvel;
  const float wvno2 = wvno * wvno;
  const float oinv  = 1.0f / omega;
  const float oinv2 = oinv * oinv;

  // ---- halfspace seed vector e[0..4] ----
  {
    // scoped so the loop body can reuse names
  }
  float xka = omega * sAinv[NL - 1];
  float xkb = omega * sBinv[NL - 1];
  float ra  = fast_sqrt((wvno + xka) * fabsf(wvno - xka));
  float rb  = fast_sqrt((wvno + xkb) * fabsf(wvno - xkb));
  float gammk = sB2[NL - 1] * oinv2;
  float gam   = gammk * wvno2;
  float gamm1 = gam - 1.0f;
  float r     = sRho[NL - 1];

  float e0 = r * r * (gamm1 * gamm1 - gam * gammk * ra * rb);
  float e1 = -r * ra;
  float e2 = r * (gamm1 - gammk * ra * rb);
  float e3 = r * rb;
  float e4 = wvno2 - ra * rb;

  // ---- reverse scan over layers NL-2 .. 0 ----
  for (int m = NL - 2; m >= 0; --m) {
    const float dm    = sD[m];
    const float rm    = sRho[m];
    const float rminv = sRinv[m];
    const float r2inv = sR2inv[m];

    xka = omega * sAinv[m];
    xkb = omega * sBinv[m];
    ra  = fast_sqrt((wvno + xka) * fabsf(wvno - xka));
    rb  = fast_sqrt((wvno + xkb) * fabsf(wvno - xkb));
    gammk = sB2[m] * oinv2;
    gam   = gammk * wvno2;

    const float p = ra * dm;
    const float q = rb * dm;

    // ---- _var ----
    const float ra_inv = fast_rcp(ra > 0.0f ? ra : 1.0f);
    const float rb_inv = fast_rcp(rb > 0.0f ? rb : 1.0f);

    const float sinp   = __sinf(p);
    const float facp   = (p < 16.0f) ? __expf(-2.0f * p) : 0.0f;
    const float sinp_e = (1.0f - facp) * 0.5f;
    const bool  ltp = wvno < xka;
    const bool  eqp = wvno == xka;
    const float pex  = (ltp || eqp) ? 0.0f : p;
    const float cosp = ltp ? __cosf(p) : (eqp ? 1.0f : (1.0f + facp) * 0.5f);
    const float w = eqp ? dm  : (ltp ? sinp : sinp_e) * ra_inv;
    const float x = eqp ? 0.f : (ltp ? -ra * sinp : ra * sinp_e);

    const float sinq   = __sinf(q);
    const float facq   = (q < 16.0f) ? __expf(-2.0f * q) : 0.0f;
    const float sinq_e = (1.0f - facq) * 0.5f;
    const bool  lts = wvno < xkb;
    const bool  eqs = wvno == xkb;
    const float sex  = (lts || eqs) ? 0.0f : q;
    const float cosq = lts ? __cosf(q) : (eqs ? 1.0f : (1.0f + facq) * 0.5f);
    const float y = eqs ? dm  : (lts ? sinq : sinq_e) * rb_inv;
    const float z = eqs ? 0.f : (lts ? -rb * sinq : rb * sinq_e);

    const float exa = pex + sex;
    const float a0  = (exa < 60.0f) ? __expf(-exa) : 0.0f;

    const float cpcq = cosp * cosq;
    const float cpy  = cosp * y;
    const float cpz  = cosp * z;
    const float cqw  = cosq * w;
    const float cqx  = cosq * x;
    const float xy   = x * y;
    const float xz   = x * z;
    const float wy   = w * y;
    const float wz   = w * z;

    gamm1 = gam - 1.0f;
    const float twgm1 = gam + gamm1;
    const float gmgmk = gam * gammk;
    const float gmgm1 = gam * gamm1;
    const float gm1sq = gamm1 * gamm1;
    const float rho2  = rm * rm;
    const float a0pq  = a0 - cpcq;
    const float t2    = -2.0f * wvno2;

    const float c00 = cpcq - 2.0f * gmgm1 * a0pq - gmgmk * xz - wvno2 * gm1sq * wy;
    const float c01 = (wvno2 * cpy - cqx) * rminv;
    const float c02 = -(twgm1 * a0pq + gammk * xz + wvno2 * gamm1 * wy) * rminv;
    const float c03 = (cpz - wvno2 * cqw) * rminv;
    const float c04 = -(2.0f * wvno2 * a0pq + xz + wvno2 * wvno2 * wy) * r2inv;
    const float c10 = (gmgmk * cpz - gm1sq * cqw) * rm;
    const float c11 = cpcq;
    const float c12 = gammk * cpz - gamm1 * cqw;
    const float c13 = -wz;
    const float c14 = c03;
    const float c30 = (gm1sq * cpy - gmgmk * cqx) * rm;
    const float c31 = -xy;
    const float c32 = gamm1 * cpy - gammk * cqx;
    const float c33 = cpcq;
    const float c34 = c01;
    const float c40 = -(2.0f * gmgmk * gm1sq * a0pq + gmgmk * gmgmk * xz
                        + gm1sq * gm1sq * wy) * rho2;
    const float c41 = c30;
    const float c42 = -(gmgm1 * twgm1 * a0pq + gam * gammk * gammk * xz
                        + gamm1 * gm1sq * wy) * rm;
    const float c43 = c10;
    const float c44 = c00;
    const float c20 = t2 * c42;
    const float c21 = t2 * c32;
    const float c22 = a0 + 2.0f * (cpcq - c00);
    const float c23 = t2 * c12;
    const float c24 = t2 * c02;

    // ee_i = sum_j e_j * ca[j][i]
    float ee0 = fmaf(e0, c00, fmaf(e1, c10, fmaf(e2, c20, fmaf(e3, c30, e4 * c40))));
    float ee1 = fmaf(e0, c01, fmaf(e1, c11, fmaf(e2, c21, fmaf(e3, c31, e4 * c41))));
    float ee2 = fmaf(e0, c02, fmaf(e1, c12, fmaf(e2, c22, fmaf(e3, c32, e4 * c42))));
    float ee3 = fmaf(e0, c03, fmaf(e1, c13, fmaf(e2, c23, fmaf(e3, c33, e4 * c43))));
    float ee4 = fmaf(e0, c04, fmaf(e1, c14, fmaf(e2, c24, fmaf(e3, c34, e4 * c44))));

    float nrm = fmaxf(fmaxf(fabsf(ee0), fabsf(ee1)),
                      fmaxf(fabsf(ee2), fmaxf(fabsf(ee3), fabsf(ee4))));
    nrm = (nrm < 1e-30f) ? 1.0f : nrm;
    const float ninv = fast_rcp(nrm);
    e0 = ee0 * ninv; e1 = ee1 * ninv; e2 = ee2 * ninv;
    e3 = ee3 * ninv; e4 = ee4 * ninv;
  }
  return e0;
}

// ---------------------------------------------------------------------------
// 1) precompute omega-independent per-layer constants (Brocher relations)
//    prep layout: [0]=d [1]=1/a [2]=1/b [3]=2b^2 [4]=rho [5]=1/rho [6]=1/rho^2
// ---------------------------------------------------------------------------
__global__ void prep_kernel(const float* __restrict__ d,
                            const float* __restrict__ b,
                            float* __restrict__ prep) {
  int i = threadIdx.x;
  if (i >= NL) return;
  float bb = b[i];
  float b2 = bb * bb;
  float a  = 0.9409f + 2.0947f * bb - 0.8206f * b2 + 0.2683f * b2 * bb
           - 0.0251f * b2 * b2;
  float a2 = a * a;
  float rho = 1.6612f * a - 0.4721f * a2 + 0.0671f * a2 * a
            - 0.0043f * a2 * a2 + 0.000106f * a2 * a2 * a;
  prep[0 * NL + i] = d[i];
  prep[1 * NL + i] = 1.0f / a;
  prep[2 * NL + i] = 1.0f / bb;
  prep[3 * NL + i] = 2.0f * b2;
  prep[4 * NL + i] = rho;
  prep[5 * NL + i] = 1.0f / rho;
  prep[6 * NL + i] = 1.0f / (rho * rho);
}

// ---------------------------------------------------------------------------
// 2) init per-T extrema
// ---------------------------------------------------------------------------
__global__ void init_kernel(unsigned* __restrict__ enc_max,
                            unsigned* __restrict__ enc_min) {
  int i = blockIdx.x * blockDim.x + threadIdx.x;
  if (i < NOBS) { enc_max[i] = 0u; enc_min[i] = 0xFFFFFFFFu; }
}

// ---------------------------------------------------------------------------
// 3) determinant grid: block = (one T, 256 C values). omega uniform per block.
// ---------------------------------------------------------------------------
__global__ void det_grid_kernel(const float* __restrict__ tlist,
                                const float* __restrict__ Clist,
                                const float* __restrict__ prep,
                                unsigned* __restrict__ enc_max,
                                unsigned* __restrict__ enc_min) {
  __shared__ float sL[7 * NL];
  __shared__ float red[256];
  const int tid = threadIdx.x;
  if (tid < NL) {
#pragma unroll
    for (int k = 0; k < 7; ++k) sL[k * NL + tid] = prep[k * NL + tid];
  }
  const int   t     = blockIdx.x;
  const float omega = fmaxf(TWO_PI_F / tlist[t], 1e-4f);
  const int   ci    = blockIdx.y * blockDim.x + tid;
  const bool  act   = ci < NC;
  const float c     = Clist[act ? ci : 0];
  __syncthreads();

  const float det = dltar4_dev(c, omega, sL, sL + NL, sL + 2 * NL, sL + 3 * NL,
                               sL + 4 * NL, sL + 5 * NL, sL + 6 * NL);

  // block max
  red[tid] = act ? det : -3.4028235e38f;
  __syncthreads();
  for (int s = 128; s > 0; s >>= 1) {
    if (tid < s) red[tid] = fmaxf(red[tid], red[tid + s]);
    __syncthreads();
  }
  const float bmax = red[0];
  __syncthreads();
  // block min
  red[tid] = act ? det : 3.4028235e38f;
  __syncthreads();
  for (int s = 128; s > 0; s >>= 1) {
    if (tid < s) red[tid] = fminf(red[tid], red[tid + s]);
    __syncthreads();
  }
  if (tid == 0) {
    atomicMax(&enc_max[t], enc_f(bmax));
    atomicMin(&enc_min[t], enc_f(red[0]));
  }
}

// ---------------------------------------------------------------------------
// 4) per-observation determinant, scaled by rng, mapped through 0.1^|x|-1
// ---------------------------------------------------------------------------
__global__ void obs_kernel(const float* __restrict__ vlist,
                           const float* __restrict__ tlist,
                           const float* __restrict__ prep,
                           const unsigned* __restrict__ enc_max,
                           const unsigned* __restrict__ enc_min,
                           float* __restrict__ obs) {
  int j = blockIdx.x * blockDim.x + threadIdx.x;
  if (j >= NOBS) return;
  const float omega = fmaxf(TWO_PI_F / tlist[j], 1e-4f);
  const float det = dltar4_dev(vlist[j], omega, prep, prep + NL, prep + 2 * NL,
                               prep + 3 * NL, prep + 4 * NL, prep + 5 * NL,
                               prep + 6 * NL);
  const float rng = dec_f(enc_max[j]) - dec_f(enc_min[j]);
  const float e   = det / rng;
  const float v   = __expf(LN01_F * fabsf(e)) - 1.0f;  // 0.1^|e| - 1
  obs[j] = fabsf(v);
}

// ---------------------------------------------------------------------------
// 5) Laplacian regularizer  y = L @ b  via V_WMMA_F32_16X16X4_F32.
//    One wave (32 lanes, EXEC all ones). For each 16-row tile of L we chain
//    16 accumulating WMMAs over K-chunks of 4 columns; b sits in column 0 of
//    the 4x16 B operand, so D[:,0] accumulates the matvec.
// ---------------------------------------------------------------------------
__device__ __forceinline__ float Lentry(int i, int j) {
  if (i == j)               return (i == 0 || i == NL - 1) ? 1.0f : 2.0f;
  if (i - j == 1 || j - i == 1) return -1.0f;
  return 0.0f;
}

__global__ void lap_wmma_kernel(const float* __restrict__ b,
                                float* __restrict__ lap) {
  __shared__ float bs[NL];
  const int lane = threadIdx.x;        // 0..31, one full wave32
  bs[lane]      = b[lane];
  bs[lane + 32] = b[lane + 32];
  __syncthreads();

  const int  mrow = lane & 15;
  const bool hi   = lane >= 16;

  for (int rt = 0; rt < 4; ++rt) {
    v8f acc = {0.f, 0.f, 0.f, 0.f, 0.f, 0.f, 0.f, 0.f};
    const int M = rt * 16 + mrow;
    for (int kk = 0; kk < 16; ++kk) {
      // A (16x4): lanes 0-15 hold K=0,1 in vgpr0,1; lanes 16-31 hold K=2,3.
      const int k0 = 4 * kk + (hi ? 2 : 0);
      v2f A;
      A.x = Lentry(M, k0);
      A.y = Lentry(M, k0 + 1);
      // B (4x16): vgpr = K&1, lane = N + 16*(K>=2). Column N=0 carries b.
      v2f B;
      B.x = (mrow == 0) ? bs[4 * kk + (hi ? 2 : 0)] : 0.0f;
      B.y = (mrow == 0) ? bs[4 * kk + (hi ? 3 : 1)] : 0.0f;
      acc = __builtin_amdgcn_wmma_f32_16x16x4_f32(
          /*neg_a=*/false, A, /*neg_b=*/false, B,
          /*c_mod=*/(short)0, acc, /*reuse_a=*/false, /*reuse_b=*/false);
    }
    // D layout: vgpr r, lane 0 -> (M=r, N=0); lane 16 -> (M=r+8, N=0)
    if (mrow == 0) {
#pragma unroll
      for (int i = 0; i < 8; ++i) {
        lap[rt * 16 + (hi ? 8 : 0) + i] = acc[i];
      }
    }
  }
}

// ---------------------------------------------------------------------------
// 6) deterministic final reduction -> scalar loss
// ---------------------------------------------------------------------------
__global__ void final_kernel(const float* __restrict__ obs,
                             const float* __restrict__ lap,
                             float* __restrict__ out) {
  __shared__ float s[256];
  const int tid = threadIdx.x;
  float v = obs[tid];
  if (tid + 256 < NOBS) v += obs[tid + 256];
  s[tid] = v;
  __syncthreads();
  for (int st = 128; st > 0; st >>= 1) {
    if (tid < st) s[tid] += s[tid + st];
    __syncthreads();
  }
  const float obs_sum = s[0];
  __syncthreads();
  s[tid] = (tid < NL) ? fabsf(lap[tid]) : 0.0f;
  __syncthreads();
  for (int st = 128; st > 0; st >>= 1) {
    if (tid < st) s[tid] += s[tid + st];
    __syncthreads();
  }
  if (tid == 0) {
    out[0] = obs_sum * (1.0f / (float)NOBS) + s[0] * (DAMP / (float)NL);
  }
}

// ---------------------------------------------------------------------------
// entry point
// ---------------------------------------------------------------------------
extern "C" void kernel_launch(void* const* d_in, const int* in_sizes, int n_in,
                              void* d_out, int out_size, void* d_ws,
                              size_t ws_size, hipStream_t stream) {
  (void)in_sizes; (void)n_in; (void)out_size; (void)ws_size;
  const float* vlist = (const float*)d_in[0];
  const float* tlist = (const float*)d_in[1];
  const float* d     = (const float*)d_in[2];
  const float* b     = (const float*)d_in[3];
  const float* Clist = (const float*)d_in[4];
  float* out = (float*)d_out;

  // workspace layout (all 4-byte typed)
  unsigned* enc_max = (unsigned*)d_ws;          // [NOBS]
  unsigned* enc_min = enc_max + NOBS;           // [NOBS]
  float*    obs     = (float*)(enc_min + NOBS); // [NOBS]
  float*    lap     = obs + NOBS;               // [NL]
  float*    prep    = lap + NL;                 // [7*NL]

  prep_kernel<<<1, NL, 0, stream>>>(d, b, prep);
  init_kernel<<<(NOBS + 255) / 256, 256, 0, stream>>>(enc_max, enc_min);
  det_grid_kernel<<<dim3(NOBS, (NC + 255) / 256), 256, 0, stream>>>(
      tlist, Clist, prep, enc_max, enc_min);
  obs_kernel<<<(NOBS + 255) / 256, 256, 0, stream>>>(vlist, tlist, prep,
                                                     enc_max, enc_min, obs);
  lap_wmma_kernel<<<1, 32, 0, stream>>>(b, lap);
  final_kernel<<<1, 256, 0, stream>>>(obs, lap, out);
}